// MyModel_86732569575842
// MI455X (gfx1250) — compile-verified
//
#include <hip/hip_runtime.h>

// Greedy NMS (RetinaFace-style) for MI455X / gfx1250.
// N = 43008 priors, MAX_DET = 512, threshold 0.4. Latency-bound serial scan:
// single persistent workgroup (1024 thr = 32 wave32 waves on one WGP).
//  - live scores: 42/thread, VGPR-resident (zero memory traffic for updates)
//  - first 8192 boxes async-staged to LDS (ASYNCcnt path), rest stream from L2
//  - argmax: wave32 shuffles + double-buffered 32-partial reduce -> 1 barrier/step
//  - suppression folded to iw*(3.5*ih) > areaI+areaJ (== iou > 0.4), self-suppressing
//  - uniform early-exit when the live set is exhausted (rows stay zero-padded)

#define TPB    1024
#define ITEMS  42          // 43008 / 1024
#define MAXDET 512
#define KLDS   8           // first KLDS*TPB = 8192 boxes cached in LDS (128 KB)
#define NEGF   (-1e30f)

__global__ __launch_bounds__(TPB, 1)
void nms_retina_kernel(const float* __restrict__ loc,
                       const float* __restrict__ conf,
                       const float* __restrict__ landms,
                       float* __restrict__ out)
{
    __shared__ float4 s_box[KLDS * TPB];   // 128 KB box cache
    __shared__ float  s_red[2][32];        // double-buffered partial maxima
    __shared__ int    s_redi[2][32];
    __shared__ int    s_outi[MAXDET];
    __shared__ float  s_outv[MAXDET];

    const int t = threadIdx.x;
    const float4* __restrict__ gbox = (const float4*)loc;

    // ---- async-stage first 8192 boxes into LDS (gfx1250 ASYNCcnt path) ----
    unsigned lbase = (unsigned)(unsigned long long)(void*)&s_box[0];
    #pragma unroll
    for (int k = 0; k < KLDS; ++k) {
        int j = k * TPB + t;
        unsigned long long ga = (unsigned long long)(gbox + j);
        unsigned la = lbase + (unsigned)j * 16u;
        asm volatile("global_load_async_to_lds_b128 %0, %1, off"
                     :: "v"(la), "v"(ga) : "memory");
    }

    // zero-pad the detection list (early-exit leaves tail rows at v=0, idx=0)
    if (t < MAXDET) { s_outi[t] = 0; s_outv[t] = 0.0f; }

    // ---- live scores -> VGPRs ----
    float sc[ITEMS];
    #pragma unroll
    for (int k = 0; k < ITEMS; ++k)
        sc[k] = conf[(k * TPB + t) * 2 + 1];

    // ---- initial local argmax (ascending j => first occurrence wins) ----
    float bs = -INFINITY; int bi = 0x7fffffff;
    #pragma unroll
    for (int k = 0; k < ITEMS; ++k) {
        int j = k * TPB + t;
        if (sc[k] > bs) { bs = sc[k]; bi = j; }
    }

    asm volatile("s_wait_asynccnt 0" ::: "memory");
    __syncthreads();                        // LDS box cache visible to all waves

    const int wave = t >> 5;
    const int lane = t & 31;

    #pragma unroll 1
    for (int it = 0; it < MAXDET; ++it) {
        // ---- intra-wave argmax (wave32 shuffles) ----
        #pragma unroll
        for (int off = 16; off > 0; off >>= 1) {
            float os = __shfl_xor(bs, off, 32);
            int   oi = __shfl_xor(bi, off, 32);
            if (os > bs || (os == bs && oi < bi)) { bs = os; bi = oi; }
        }
        const int p = it & 1;
        if (lane == 0) { s_red[p][wave] = bs; s_redi[p][wave] = bi; }
        __syncthreads();                    // the only barrier per step

        // ---- every wave redundantly reduces the 32 partials ----
        float rs = s_red[p][lane];
        int   ri = s_redi[p][lane];
        #pragma unroll
        for (int off = 16; off > 0; off >>= 1) {
            float os = __shfl_xor(rs, off, 32);
            int   oi = __shfl_xor(ri, off, 32);
            if (os > rs || (os == rs && oi < ri)) { rs = os; ri = oi; }
        }
        const int   sel = __builtin_amdgcn_readfirstlane(ri);
        const float rsu = __int_as_float(
            __builtin_amdgcn_readfirstlane(__float_as_int(rs)));

        if (!(rsu > NEGF * 0.5f)) break;    // live set exhausted: uniform exit

        const float4 b = gbox[sel];         // uniform address -> scalar load
        const float  areaI = (b.z - b.x) * (b.w - b.y);
        if (t == 0) { s_outi[it] = sel; s_outv[it] = 1.0f; }

        __builtin_prefetch(gbox + KLDS * TPB + t, 0, 0);   // global_prefetch_b8

        // ---- fused suppression + next local argmax ----
        // iou > 0.4 <=> inter > 0.4*(aI+aJ-inter) <=> iw*(3.5*ih) > aI+aJ
        // (3.5 folded through the positive-scale-invariant max clamp)
        // selected box self-suppresses (iou(self)=1), so no j==sel special case
        bs = -INFINITY; bi = 0x7fffffff;
        #pragma unroll
        for (int k = 0; k < ITEMS; ++k) {
            int j = k * TPB + t;
            float4 q = (k < KLDS) ? s_box[j] : gbox[j];
            float xx1 = fmaxf(b.x, q.x), yy1 = fmaxf(b.y, q.y);
            float xx2 = fminf(b.z, q.z), yy2 = fminf(b.w, q.w);
            float iw   = fmaxf(xx2 - xx1, 0.0f);
            float ih35 = fmaxf((yy2 - yy1) * 3.5f, 0.0f);
            float sum  = fmaf(q.z - q.x, q.w - q.y, areaI);   // aI + aJ
            float s = (iw * ih35 > sum) ? NEGF : sc[k];
            sc[k] = s;
            if (s > bs) { bs = s; bi = j; }
        }
        // no trailing barrier: next step writes the *other* s_red buffer, and
        // buffer p is only rewritten after everyone passed the next barrier.
    }

    __syncthreads();

    // ---- gather outputs: [boxes 512x4 | landms 512x10 | scores 512] ----
    if (t < MAXDET) {
        int   i = s_outi[t];
        float v = s_outv[t];
        float4 b = gbox[i];
        out[t * 4 + 0] = b.x * v;
        out[t * 4 + 1] = b.y * v;
        out[t * 4 + 2] = b.z * v;
        out[t * 4 + 3] = b.w * v;
        const float* lm = landms + (size_t)i * 10;
        float* lo = out + MAXDET * 4 + t * 10;
        #pragma unroll
        for (int q2 = 0; q2 < 10; ++q2) lo[q2] = lm[q2] * v;
        out[MAXDET * 4 + MAXDET * 10 + t] = conf[2 * i + 1] * v;
    }
}

extern "C" void kernel_launch(void* const* d_in, const int* in_sizes, int n_in,
                              void* d_out, int out_size, void* d_ws, size_t ws_size,
                              hipStream_t stream) {
    (void)in_sizes; (void)n_in; (void)out_size; (void)d_ws; (void)ws_size;
    const float* loc    = (const float*)d_in[0];
    const float* conf   = (const float*)d_in[1];
    const float* landms = (const float*)d_in[2];
    float* out = (float*)d_out;
    hipLaunchKernelGGL(nms_retina_kernel, dim3(1), dim3(TPB), 0, stream,
                       loc, conf, landms, out);
}